// Static_Influence_Model_86449101734280
// MI455X (gfx1250) — compile-verified
//
#include <hip/hip_runtime.h>

// Problem constants (match reference)
#define T_ 4
#define N_ 100000
#define G_ 128
#define R_ 4
#define B_ 4096
#define K_ 32
#define I_ 128

typedef float v2f __attribute__((ext_vector_type(2)));
typedef float v8f __attribute__((ext_vector_type(8)));

// LDS A tile: 16 rows x 512 K-values, padded stride for bank-conflict-free
// WMMA A-fragment reads (row*516 -> bank rotation of 4 per row).
#define LDS_STRIDE 516

__global__ __launch_bounds__(256) void sim_fused_kernel(
    const float* __restrict__ emb,          // [T, N, G]
    const float* __restrict__ wts,          // [R, G, I] == [512, 128]
    const int* __restrict__ nidx,           // [T, R, B, K]
    const unsigned char* __restrict__ nmask,// [T, R, B, K] (bool, 1 byte)
    float* __restrict__ out)                // [T*B, I]
{
    __shared__ float A_s[16 * LDS_STRIDE];  // 33,024 B

    const int lane = threadIdx.x & 31;
    const int wave = threadIdx.x >> 5;      // 0..7
    const int m0   = blockIdx.x * 16;       // first output row (m = t*B + b)
    const int t    = m0 / B_;               // B_ multiple of 16 -> block never straddles t
    const int b0   = m0 - t * B_;

    // ---------------- Phase 1: masked-mean gather into LDS ----------------
    // 64 (row, r) tasks per block; each wave handles 8.
    #pragma unroll 2
    for (int j = 0; j < 8; ++j) {
        const int task = wave * 8 + j;      // 0..63
        const int row  = task >> 2;         // 0..15
        const int r    = task & 3;          // 0..3
        const int b    = b0 + row;
        const long base = (((long)t * R_ + r) * B_ + b) * K_;
        const int g4 = lane * 4;            // lane covers g4..g4+3 of G=128

        float4 acc = make_float4(0.f, 0.f, 0.f, 0.f);
        float cnt = 0.f;
        for (int k = 0; k < K_; ++k) {
            const int nid = nidx[base + k];            // wave-uniform
            const unsigned char mk = nmask[base + k];  // wave-uniform
            if (mk) {                                   // uniform branch: skip dead gathers
                const float4 e = *(const float4*)(emb + ((long)t * N_ + nid) * G_ + g4);
                acc.x += e.x; acc.y += e.y; acc.z += e.z; acc.w += e.w;
                cnt += 1.f;
            }
        }
        const float inv = 1.f / fmaxf(cnt, 1.f);
        float4 mean = make_float4(acc.x * inv, acc.y * inv, acc.z * inv, acc.w * inv);
        // A[row][r*128 + g] layout: contraction index kk = r*G + g
        *(float4*)(&A_s[row * LDS_STRIDE + r * G_ + g4]) = mean;
    }

    __syncthreads();

    // ------------- Phase 2: C[16x128] = A[16x512] x W[512x128] -------------
    // Each wave owns one 16-wide N tile. f32 WMMA: 16x16x4.
    const int n0   = wave * 16;
    const int half = lane >> 4;   // 0 = lanes 0-15 (K pair 0,1), 1 = lanes 16-31 (K pair 2,3)
    const int l15  = lane & 15;

    v8f c = {};
    #pragma unroll 4
    for (int k0 = 0; k0 < R_ * G_; k0 += 4) {
        const int kb = k0 + half * 2;
        // A fragment (ISA 32-bit 16x4 layout): lane half selects K pair, M = lane&15
        v2f a = *(const v2f*)(&A_s[l15 * LDS_STRIDE + kb]);
        // B fragment (mirror layout): row kb / kb+1, column n0 + (lane&15)
        v2f b;
        b.x = wts[(long)kb * I_ + n0 + l15];
        b.y = wts[((long)kb + 1) * I_ + n0 + l15];
        c = __builtin_amdgcn_wmma_f32_16x16x4_f32(
            /*neg_a=*/false, a, /*neg_b=*/false, b,
            /*c_mod=*/(short)0, c, /*reuse_a=*/false, /*reuse_b=*/false);
    }

    // Store C per ISA 16x16 f32 C/D layout:
    //   VGPR v: lanes 0-15 -> M = v, N = lane; lanes 16-31 -> M = v+8, N = lane-16
    const int col = n0 + l15;
    #pragma unroll
    for (int v = 0; v < 8; ++v) {
        const int m = m0 + v + half * 8;
        out[(long)m * I_ + col] = c[v];
    }
}

extern "C" void kernel_launch(void* const* d_in, const int* in_sizes, int n_in,
                              void* d_out, int out_size, void* d_ws, size_t ws_size,
                              hipStream_t stream) {
    (void)in_sizes; (void)n_in; (void)out_size; (void)d_ws; (void)ws_size;
    const float* emb          = (const float*)d_in[0];
    const float* wts          = (const float*)d_in[1];
    const int* nidx           = (const int*)d_in[2];
    const unsigned char* nmsk = (const unsigned char*)d_in[3];
    // d_in[4] = train_year scalar (T_), compile-time constant here.
    float* out = (float*)d_out;

    dim3 grid((T_ * B_) / 16);  // 1024 blocks, one 16x128 output tile each
    sim_fused_kernel<<<grid, 256, 0, stream>>>(emb, wts, nidx, nmsk, out);
}